// GTOElectrostaticEnergy_80109730005617
// MI455X (gfx1250) — compile-verified
//
#include <hip/hip_runtime.h>
#include <hip/hip_bf16.h>

// ---------------------------------------------------------------------------
// GTO / Ewald reciprocal-space electrostatic energy, fused for MI455X (gfx1250)
//
//  * Sorted k_vector_batch / batch -> only same-graph (k,node) pairs survive
//    the mask: ~16x work reduction vs dense 16384x4096.
//  * inner products k . r via V_WMMA_F32_16X16X4_F32 (K=3 padded to 4, f32).
//  * Transcendental-bound: native v_cos/v_sin/v_exp on VALU.
//  * Branch-free inner loop (clamped loads + value selects, no EXEC churn)
//    with one-chunk software pipelining so loads overlap sin/cos math.
// ---------------------------------------------------------------------------

typedef float v2f __attribute__((ext_vector_type(2)));
typedef float v8f __attribute__((ext_vector_type(8)));

__constant__ constexpr float PI_F    = 3.14159265358979323846f;
__constant__ constexpr float TWO_PI  = 6.28318530717958647692f;
__constant__ constexpr float SIGMA_F = 1.0f;
__constant__ constexpr float INV_SIGMA_SQRT_2PI = 0.39894228040143267794f; // 1/(sigma*sqrt(2*pi))

struct __attribute__((packed, aligned(4))) Float3 { float x, y, z; };

__device__ __forceinline__ int lower_bound_dev(const int* __restrict__ a, int n, int v) {
    int lo = 0, hi = n;
    while (lo < hi) {
        int mid = (lo + hi) >> 1;
        if (a[mid] < v) lo = mid + 1; else hi = mid;
    }
    return lo;
}

// ---------------------------------------------------------------------------
// Kernel 0: per-graph segment boundaries, tile-offset prefix, accumulator init
// ---------------------------------------------------------------------------
__global__ void seg_init_kernel(const int* __restrict__ kvb, const int* __restrict__ batch,
                                int* __restrict__ kseg, int* __restrict__ nseg,
                                int* __restrict__ tileoff,
                                float* __restrict__ eacc, float* __restrict__ sacc,
                                int K, int N, int B) {
    const int t = threadIdx.x;
    if (t <= B)                   kseg[t]      = lower_bound_dev(kvb,   K, t);
    if (t >= 64 && (t - 64) <= B) nseg[t - 64] = lower_bound_dev(batch, N, t - 64);
    if (t < B) { eacc[t] = 0.0f; sacc[t] = 0.0f; }
    __syncthreads();
    if (t == 0) {
        tileoff[0] = 0;
        for (int b = 0; b < B; ++b) {
            int cnt = (kseg[b + 1] - kseg[b] + 15) >> 4;   // 16-k tiles in graph b
            tileoff[b + 1] = tileoff[b] + cnt;
        }
    }
}

// ---------------------------------------------------------------------------
// Kernel 1: main reciprocal-space sum. One wave32 per 16-k tile of one graph.
// ---------------------------------------------------------------------------
__global__ __launch_bounds__(32)
void ewald_recip_kernel(const float* __restrict__ kvec, const float* __restrict__ kn2,
                        const float* __restrict__ k0m,  const float* __restrict__ q,
                        const float* __restrict__ pos,  const float* __restrict__ vol,
                        const int* __restrict__ kseg,   const int* __restrict__ nseg,
                        const int* __restrict__ tileoff,
                        float* __restrict__ eacc, int K, int N, int B) {
    const int t = blockIdx.x;
    if (t >= tileoff[B]) return;                       // wave-uniform exit
    int b = 0;
    while (t >= tileoff[b + 1]) ++b;                   // B=16: tiny scalar scan
    const int tile  = t - tileoff[b];
    const int ks    = kseg[b], ke = kseg[b + 1];
    const int kbase = ks + tile * 16;
    const int ns    = nseg[b], ne = nseg[b + 1];

    const int  lane   = threadIdx.x;   // 0..31, wave32
    const int  m      = lane & 15;
    const bool hiHalf = lane >= 16;

    const Float3* __restrict__ pos3 = (const Float3*)pos;

    // A fragment: 16x4 f32 (rows = 16 k-vectors, cols = kx,ky,kz,0)
    // lanes 0-15: VGPR0=K0(kx), VGPR1=K1(ky); lanes 16-31: VGPR0=K2(kz), VGPR1=K3(0)
    {
    }
    int kidx = kbase + m; if (kidx >= K) kidx = K - 1;
    const Float3 kv = ((const Float3*)kvec)[kidx];
    v2f afrag;
    afrag.x = hiHalf ? kv.z : kv.x;
    afrag.y = hiHalf ? 0.0f : kv.y;

    float accC[8], accS[8];
#pragma unroll
    for (int v = 0; v < 8; ++v) { accC[v] = 0.0f; accS[v] = 0.0f; }

    if (ns < ne) {
        // ---- software-pipelined node loop: loads of chunk i+1 overlap the
        //      sin/cos burst of chunk i. All loads unconditional (clamped to
        //      < N, garbage columns masked by qn = 0 at accumulate time).
        int ni0 = ns + m; if (ni0 >= N) ni0 = N - 1;
        Float3 p  = pos3[ni0];
        float  qv = q[ni0];

        for (int nb = ns; nb < ne; nb += 16) {
            // Current chunk fragments (value selects -> v_cndmask, no branches)
            v2f bfrag;
            bfrag.x = hiHalf ? p.z  : p.x;
            bfrag.y = hiHalf ? 0.0f : p.y;
            const float qn = ((nb + m) < ne) ? qv : 0.0f;

            // Prefetch next chunk (uniform branch; EXEC stays all-1s)
            const int nnext = nb + 16;
            if (nnext < ne) {
                int ni = nnext + m; if (ni >= N) ni = N - 1;
                p  = pos3[ni];
                qv = q[ni];
            }

            v8f c = {0.f, 0.f, 0.f, 0.f, 0.f, 0.f, 0.f, 0.f};
            c = __builtin_amdgcn_wmma_f32_16x16x4_f32(
                    /*neg_a=*/false, afrag, /*neg_b=*/false, bfrag,
                    /*c_mod=*/(short)0, c, /*reuse_a=*/false, /*reuse_b=*/false);

            // lane holds rows v (+8 for hi half), column m of the 16x16 tile
#pragma unroll
            for (int v = 0; v < 8; ++v) {
                const float ang = c[v];
                accC[v] = __builtin_fmaf(__cosf(ang), qn, accC[v]);
                accS[v] = __builtin_fmaf(__sinf(ang), qn, accS[v]);
            }
        }
    }

    // Reduce over the 16 columns held by each half-wave (xor butterflies stay
    // inside each 16-lane group).
#pragma unroll
    for (int v = 0; v < 8; ++v) {
#pragma unroll
        for (int mk = 1; mk < 16; mk <<= 1) {
            accC[v] += __shfl_xor(accC[v], mk, 32);
            accS[v] += __shfl_xor(accS[v], mk, 32);
        }
    }

    // Lane 0 finalizes rows 0..7 (k = kbase..kbase+7); lane 16 rows 8..15.
    if (m == 0) {
        const int   rowoff    = hiHalf ? 8 : 0;
        const float volb      = vol[b];
        const float prefscale = (TWO_PI * TWO_PI * TWO_PI) / volb;
        float partial = 0.0f;
#pragma unroll
        for (int v = 0; v < 8; ++v) {
            const int kk = kbase + rowoff + v;
            if (kk < ke) {                               // drop padded rows
                const float kn   = kn2[kk];
                const float fs   = __expf(-0.5f * SIGMA_F * SIGMA_F * kn);
                const float pref = fs * prefscale;
                const float kf   = (k0m[kk] > 0.0f) ? 0.0f : (1.0f / kn);
                const float dc   = accC[v] * pref;
                const float dsv  = accS[v] * pref;
                partial += 8.0f * PI_F * kf * (dc * dc + dsv * dsv);
            }
        }
        atomicAdd(&eacc[b], partial);
    }
}

// ---------------------------------------------------------------------------
// Kernel 2: self-energy  sum_b q^2 / (sigma*sqrt(2 pi))
// ---------------------------------------------------------------------------
__global__ void self_energy_kernel(const float* __restrict__ q, const int* __restrict__ batch,
                                   float* __restrict__ sacc, int N) {
    const int i = blockIdx.x * blockDim.x + threadIdx.x;
    if (i < N) {
        const float qv = q[i];
        atomicAdd(&sacc[batch[i]], qv * qv * INV_SIGMA_SQRT_2PI);
    }
}

// ---------------------------------------------------------------------------
// Kernel 3: energy[b] = 0.5*vol*eacc/(2pi)^6 - 0.5*self
// ---------------------------------------------------------------------------
__global__ void finalize_kernel(const float* __restrict__ vol, const float* __restrict__ eacc,
                                const float* __restrict__ sacc, float* __restrict__ out, int B) {
    const int b = threadIdx.x;
    if (b < B) {
        constexpr float inv2pi6 =
            1.0f / (TWO_PI * TWO_PI * TWO_PI * TWO_PI * TWO_PI * TWO_PI);
        out[b] = 0.5f * vol[b] * eacc[b] * inv2pi6 - 0.5f * sacc[b];
    }
}

// ---------------------------------------------------------------------------
extern "C" void kernel_launch(void* const* d_in, const int* in_sizes, int n_in,
                              void* d_out, int out_size, void* d_ws, size_t ws_size,
                              hipStream_t stream) {
    const float* kvec  = (const float*)d_in[0];  // (K,3)
    const float* kn2   = (const float*)d_in[1];  // (K,)
    const int*   kvb   = (const int*)  d_in[2];  // (K,) sorted
    const float* k0m   = (const float*)d_in[3];  // (K,)
    const float* q     = (const float*)d_in[4];  // (N,1)
    const float* pos   = (const float*)d_in[5];  // (N,3)
    const int*   batch = (const int*)  d_in[6];  // (N,) sorted
    const float* vol   = (const float*)d_in[7];  // (B,)
    // d_in[8] = pbc, unused

    const int K = in_sizes[1];
    const int N = in_sizes[4];
    const int B = in_sizes[7];

    int*   kseg    = (int*)d_ws;          // B+1
    int*   nseg    = kseg + (B + 1);      // B+1
    int*   tileoff = nseg + (B + 1);      // B+1
    float* eacc    = (float*)(tileoff + (B + 1));  // B
    float* sacc    = eacc + B;                      // B

    seg_init_kernel<<<1, 128, 0, stream>>>(kvb, batch, kseg, nseg, tileoff, eacc, sacc, K, N, B);

    const int maxTiles = (K + 15) / 16 + B;  // upper bound on sum of per-graph tiles
    ewald_recip_kernel<<<maxTiles, 32, 0, stream>>>(kvec, kn2, k0m, q, pos, vol,
                                                    kseg, nseg, tileoff, eacc, K, N, B);

    self_energy_kernel<<<(N + 255) / 256, 256, 0, stream>>>(q, batch, sacc, N);

    finalize_kernel<<<1, 64, 0, stream>>>(vol, eacc, sacc, (float*)d_out, B);
}